// DNA_Net_17617955848513
// MI455X (gfx1250) — compile-verified
//
#include <hip/hip_runtime.h>
#include <hip/hip_bf16.h>
#include <math.h>

#define NN     25000
#define EEDGE  400000
#define FIN    256
#define HIDD   64
#define NCLS   32

typedef __attribute__((ext_vector_type(2))) float v2f;
typedef __attribute__((ext_vector_type(8))) float v8f;

// ---------------- degree / norm ----------------
__global__ __launch_bounds__(256) void k_deg_init(float* __restrict__ dis) {
    int n = blockIdx.x * 256 + threadIdx.x;
    if (n < NN) dis[n] = 1.0f;   // self-loop contributes 1
}
__global__ __launch_bounds__(256) void k_deg_acc(const int* __restrict__ colI, float* __restrict__ dis) {
    int e = blockIdx.x * 256 + threadIdx.x;
    if (e < EEDGE) atomicAdd(&dis[colI[e]], 1.0f);
}
__global__ __launch_bounds__(256) void k_deg_fin(float* __restrict__ dis) {
    int n = blockIdx.x * 256 + threadIdx.x;
    if (n < NN) dis[n] = rsqrtf(dis[n]);  // deg >= 1 always
}

// ---------------- lin1: h = relu(x @ W1 + b1)  via f32 WMMA ----------------
// Block: 256 threads = 8 waves; block tile = 32 rows x 64 cols.
// wave w: rowTile = (w>>2)*16, colTile = (w&3)*16. K loop: 256 / 4.
__global__ __launch_bounds__(256) void k_lin1_wmma(const float* __restrict__ x,
                                                   const float* __restrict__ w1,
                                                   const float* __restrict__ b1,
                                                   float* __restrict__ xall) {
    __shared__ float xs[32][260];   // 260-float row pitch: 16B aligned, conflict-free
    const int t = threadIdx.x;
    const int rowBase = blockIdx.x * 32;
    // stage 32x256 x-tile (coalesced float4), clamp OOB rows
    for (int i = t; i < 32 * 64; i += 256) {
        int r  = i >> 6;
        int k4 = (i & 63) << 2;
        int gr = rowBase + r; if (gr >= NN) gr = NN - 1;
        const float4 v = *reinterpret_cast<const float4*>(x + (size_t)gr * FIN + k4);
        *reinterpret_cast<float4*>(&xs[r][k4]) = v;
    }
    __syncthreads();

    const int wave   = t >> 5;
    const int lane   = t & 31;
    const int rowTile = (wave >> 2) << 4;   // 0 / 16
    const int colBase = (wave & 3) << 4;    // 0 / 16 / 32 / 48
    const int m      = lane & 15;
    const int khalf  = (lane >> 4) << 1;    // 0 / 2

    v8f acc = {0.f, 0.f, 0.f, 0.f, 0.f, 0.f, 0.f, 0.f};
    const float* wcol = w1 + colBase + m;   // column of W1, stride HIDD per K
    for (int kb = 0; kb < FIN; kb += 4) {
        v2f a;
        a.x = xs[rowTile + m][kb + khalf];
        a.y = xs[rowTile + m][kb + khalf + 1];
        v2f b;
        b.x = wcol[(size_t)(kb + khalf) * HIDD];
        b.y = wcol[(size_t)(kb + khalf + 1) * HIDD];
        acc = __builtin_amdgcn_wmma_f32_16x16x4_f32(false, a, false, b, (short)0, acc, false, false);
    }
    const int col   = colBase + m;
    const float bia = b1[col];
    const int mbase = rowTile + ((lane >> 4) << 3);
    #pragma unroll
    for (int v = 0; v < 8; ++v) {
        int gn = rowBase + mbase + v;
        if (gn < NN) {
            float r = acc[v] + bia;
            xall[(size_t)gn * 256 + col] = r > 0.f ? r : 0.f;  // x_all slot 0
        }
    }
}

// ---------------- per-node grouped Q/K/V ----------------
// y[g*8+o] = b[d] + sum_i x[g*8+i] * w[l][g][i][o]
__global__ __launch_bounds__(256) void k_qkv(const float* __restrict__ xall,
                                             const float* __restrict__ wq, const float* __restrict__ bq,
                                             const float* __restrict__ wk, const float* __restrict__ bk,
                                             const float* __restrict__ wv, const float* __restrict__ bv,
                                             float* __restrict__ Qb, float* __restrict__ Kb, float* __restrict__ Vb,
                                             int l, int L) {
    const int per = (2 * L + 1) * 64;
    long long tid = (long long)blockIdx.x * 256 + threadIdx.x;
    if (tid >= (long long)NN * per) return;
    int n   = (int)(tid / per);
    int rem = (int)(tid - (long long)n * per);
    int slot = rem >> 6;
    int d    = rem & 63;
    int g = d >> 3, o = d & 7;

    const float *w, *b, *xin;
    float* out;
    if (slot == 0) {               // Q from last slot
        w = wq; b = bq; xin = xall + (size_t)n * 256 + (size_t)(L - 1) * 64; out = Qb + (size_t)n * 64;
    } else if (slot <= L) {        // K, slot-1 = ll
        int ll = slot - 1;
        w = wk; b = bk; xin = xall + (size_t)n * 256 + (size_t)ll * 64; out = Kb + ((size_t)n * 3 + ll) * 64;
    } else {                       // V
        int ll = slot - L - 1;
        w = wv; b = bv; xin = xall + (size_t)n * 256 + (size_t)ll * 64; out = Vb + ((size_t)n * 3 + ll) * 64;
    }
    const float* wp = w + ((size_t)(l * 8 + g)) * 64 + o;  // + i*8
    const float* xg = xin + g * 8;
    float acc = b[l * 64 + d];
    #pragma unroll
    for (int i = 0; i < 8; ++i) acc += xg[i] * wp[i * 8];
    out[d] = acc;
}

// ---------------- zero aggregation buffer ----------------
__global__ __launch_bounds__(256) void k_zero(float* __restrict__ p, int count) {
    int i = blockIdx.x * 256 + threadIdx.x;
    if (i < count) p[i] = 0.0f;
}

// ---------------- edge attention + scatter ----------------
// one thread per (edge, head); self-loop edges appended at e >= EEDGE
__global__ __launch_bounds__(256) void k_attn(const int* __restrict__ rowI, const int* __restrict__ colI,
                                              const float* __restrict__ dis,
                                              const float* __restrict__ Qb, const float* __restrict__ Kb,
                                              const float* __restrict__ Vb,
                                              float* __restrict__ agg, int L) {
    long long tid = (long long)blockIdx.x * 256 + threadIdx.x;
    const long long total = (long long)(EEDGE + NN) * 8;
    if (tid >= total) return;
    int e = (int)(tid >> 3);
    int h = (int)(tid & 7);

    int r, c; float nrm;
    if (e < EEDGE) {
        r = rowI[e]; c = colI[e];
        nrm = dis[r] * dis[c];
    } else {
        r = c = e - EEDGE;
        float d0 = dis[r];
        nrm = d0 * d0;
    }

    float qv[8];
    const float* qp = Qb + (size_t)c * 64 + h * 8;
    #pragma unroll
    for (int j = 0; j < 8; ++j) qv[j] = qp[j];

    float s[3];
    float mx = -3.0e38f;
    for (int ll = 0; ll < L; ++ll) {
        const float* kp = Kb + ((size_t)r * 3 + ll) * 64 + h * 8;
        float a = 0.f;
        #pragma unroll
        for (int j = 0; j < 8; ++j) a += qv[j] * kp[j];
        a *= 0.35355339059327373f;   // 1/sqrt(8)
        s[ll] = a;
        mx = fmaxf(mx, a);
    }
    float den = 0.f;
    for (int ll = 0; ll < L; ++ll) { s[ll] = expf(s[ll] - mx); den += s[ll]; }
    const float scale = nrm / den;

    float o[8] = {0.f, 0.f, 0.f, 0.f, 0.f, 0.f, 0.f, 0.f};
    for (int ll = 0; ll < L; ++ll) {
        const float wgt = s[ll];
        const float* vp = Vb + ((size_t)r * 3 + ll) * 64 + h * 8;
        #pragma unroll
        for (int j = 0; j < 8; ++j) o[j] += wgt * vp[j];
    }
    float* dst = agg + (size_t)c * 64 + h * 8;
    #pragma unroll
    for (int j = 0; j < 8; ++j) atomicAdd(&dst[j], scale * o[j]);
}

// ---------------- relu + append into x_all slot L ----------------
__global__ __launch_bounds__(256) void k_relu_store(const float* __restrict__ agg,
                                                    float* __restrict__ xall, int L) {
    int i = blockIdx.x * 256 + threadIdx.x;
    if (i >= NN * 64) return;
    int n = i >> 6, d = i & 63;
    float v = agg[i];
    xall[(size_t)n * 256 + (size_t)L * 64 + d] = v > 0.f ? v : 0.f;
}

// ---------------- lin2: logits = x_all[:,3] @ W2 + b2 via f32 WMMA ----------------
// Block: 8 waves; tile = 64 rows x 32 cols. wave: rowTile=(w>>1)*16, colTile=(w&1)*16
__global__ __launch_bounds__(256) void k_lin2_wmma(const float* __restrict__ xall,
                                                   const float* __restrict__ w2,
                                                   const float* __restrict__ b2,
                                                   float* __restrict__ logits) {
    const int t = threadIdx.x;
    const int rowBase = blockIdx.x * 64;
    const int wave = t >> 5;
    const int lane = t & 31;
    const int rowTile = (wave >> 1) << 4;   // 0..48
    const int colBase = (wave & 1) << 4;    // 0 / 16
    const int m     = lane & 15;
    const int khalf = (lane >> 4) << 1;

    int gr = rowBase + rowTile + m; if (gr >= NN) gr = NN - 1;
    const float* arow = xall + (size_t)gr * 256 + 192;   // slot 3
    const float* wcol = w2 + colBase + m;

    v8f acc = {0.f, 0.f, 0.f, 0.f, 0.f, 0.f, 0.f, 0.f};
    for (int kb = 0; kb < HIDD; kb += 4) {
        v2f a;
        a.x = arow[kb + khalf];
        a.y = arow[kb + khalf + 1];
        v2f b;
        b.x = wcol[(size_t)(kb + khalf) * NCLS];
        b.y = wcol[(size_t)(kb + khalf + 1) * NCLS];
        acc = __builtin_amdgcn_wmma_f32_16x16x4_f32(false, a, false, b, (short)0, acc, false, false);
    }
    const int col   = colBase + m;
    const float bia = b2[col];
    const int mbase = rowTile + ((lane >> 4) << 3);
    #pragma unroll
    for (int v = 0; v < 8; ++v) {
        int gn = rowBase + mbase + v;
        if (gn < NN) logits[(size_t)gn * 32 + col] = acc[v] + bia;
    }
}

// ---------------- log_softmax over 32 classes ----------------
__global__ __launch_bounds__(256) void k_logsoftmax(const float* __restrict__ logits,
                                                    float* __restrict__ out) {
    int n = blockIdx.x * 256 + threadIdx.x;
    if (n >= NN) return;
    const float* p = logits + (size_t)n * 32;
    float mx = p[0];
    #pragma unroll
    for (int c = 1; c < 32; ++c) mx = fmaxf(mx, p[c]);
    float den = 0.f;
    #pragma unroll
    for (int c = 0; c < 32; ++c) den += expf(p[c] - mx);
    const float lg = mx + logf(den);
    #pragma unroll
    for (int c = 0; c < 32; ++c) out[(size_t)n * 32 + c] = p[c] - lg;
}

extern "C" void kernel_launch(void* const* d_in, const int* in_sizes, int n_in,
                              void* d_out, int out_size, void* d_ws, size_t ws_size,
                              hipStream_t stream) {
    (void)in_sizes; (void)n_in; (void)out_size; (void)ws_size;
    const float* x  = (const float*)d_in[0];
    const int*   ei = (const int*)  d_in[1];
    const float* w1 = (const float*)d_in[2];
    const float* b1 = (const float*)d_in[3];
    const float* wq = (const float*)d_in[4];
    const float* bq = (const float*)d_in[5];
    const float* wk = (const float*)d_in[6];
    const float* bk = (const float*)d_in[7];
    const float* wv = (const float*)d_in[8];
    const float* bv = (const float*)d_in[9];
    const float* w2 = (const float*)d_in[10];
    const float* b2 = (const float*)d_in[11];
    float* out = (float*)d_out;

    // workspace carve-up (floats): total ~80.1 MB
    float* ws   = (float*)d_ws;
    float* xall = ws;                           // N * 4 * 64
    float* Qb   = xall + (size_t)NN * 256;      // N * 64
    float* Kb   = Qb   + (size_t)NN * 64;       // N * 3 * 64
    float* Vb   = Kb   + (size_t)NN * 192;      // N * 3 * 64
    float* agg  = Vb   + (size_t)NN * 192;      // N * 64
    float* dis  = agg  + (size_t)NN * 64;       // N
    float* lgt  = dis  + NN;                    // N * 32

    const int* rowI = ei;            // edge_index[0]
    const int* colI = ei + EEDGE;    // edge_index[1]

    // gcn_norm
    k_deg_init<<<(NN + 255) / 256, 256, 0, stream>>>(dis);
    k_deg_acc <<<(EEDGE + 255) / 256, 256, 0, stream>>>(colI, dis);
    k_deg_fin <<<(NN + 255) / 256, 256, 0, stream>>>(dis);

    // lin1 + relu -> x_all slot 0
    k_lin1_wmma<<<(NN + 31) / 32, 256, 0, stream>>>(x, w1, b1, xall);

    for (int l = 0; l < 3; ++l) {
        const int L = l + 1;
        const long long qkvTot = (long long)NN * (2 * L + 1) * 64;
        k_qkv<<<(unsigned)((qkvTot + 255) / 256), 256, 0, stream>>>(xall, wq, bq, wk, bk, wv, bv,
                                                                    Qb, Kb, Vb, l, L);
        k_zero<<<(NN * 64 + 255) / 256, 256, 0, stream>>>(agg, NN * 64);
        const long long attnTot = (long long)(EEDGE + NN) * 8;
        k_attn<<<(unsigned)((attnTot + 255) / 256), 256, 0, stream>>>(rowI, colI, dis, Qb, Kb, Vb, agg, L);
        k_relu_store<<<(NN * 64 + 255) / 256, 256, 0, stream>>>(agg, xall, L);
    }

    k_lin2_wmma<<<(NN + 63) / 64, 256, 0, stream>>>(xall, w2, b2, lgt);
    k_logsoftmax<<<(NN + 255) / 256, 256, 0, stream>>>(lgt, out);
}